// GCNNet_13709535609639
// MI455X (gfx1250) — compile-verified
//
#include <hip/hip_runtime.h>
#include <hip/hip_bf16.h>

// ---------------------------------------------------------------------------
// Graph Transformer (4x TransformerConv + BN/ReLU) for MI455X / gfx1250.
// Dense node GEMMs use V_WMMA_F32_16X16X4_F32 (full fp32 fidelity; GEMMs are
// ~13 GFLOP total and not the bottleneck). Edge softmax/scatter phase is
// L2-resident gather/scatter with fp32 atomics.
//
// Input flattening assumption (setup_inputs dict, recursive insertion order):
//   0: x [50000,64]   1: edge_attr [500000,1]
//   2..37:  c1..c4 each {Wq,bq,Wk,bk,Wv,bv,We,Ws,bs}
//   38..45: bn1..bn4 each {g,b}
//   46: edge_index [2,500000] int32 (row0=src, row1=dst)
// Output: x_ae [50000,64] then x_de [50000,64], fp32.
//
// Assumes N_NODES % 16 == 0 (50000 = 3125*16) -> no row-remainder guards in
// the WMMA GEMM (keeps EXEC full and loads straight-line).
// ---------------------------------------------------------------------------

typedef __attribute__((ext_vector_type(2))) float v2f;
typedef __attribute__((ext_vector_type(8))) float v8f;

#define N_NODES 50000
#define N_EDGES 500000

// ------------------------- fp32 WMMA GEMM + bias ---------------------------
// One wave computes one 16x16 tile of Y = X(Nrows x K) * W(K x NOUT) + b.
// A layout (16x4 f32): lanes 0-15 -> M=lane, VGPR0/1 = K0/K1; lanes 16-31 ->
// same rows, K2/K3.  B (4x16) mirrored.  D: VGPR r -> M=r (+8 upper half).
// K and NOUT are compile-time so all W/Y addressing is immediate offsets.
template <int K, int NOUT>
__global__ __launch_bounds__(256) void gemm_bias_wmma(
    const float* __restrict__ X, const float* __restrict__ W,
    const float* __restrict__ bias, float* __restrict__ Y, int tilesM) {
  constexpr int TILES_N = NOUT / 16;
  int wid = (blockIdx.x * blockDim.x + threadIdx.x) >> 5;
  if (wid >= tilesM * TILES_N) return;  // wave-uniform exit (EXEC stays full)
  int tm   = wid / TILES_N;
  int tn   = wid % TILES_N;
  int lane = threadIdx.x & 31;
  int mn   = lane & 15;
  int half = lane >> 4;                 // 0: K pair {0,1}, 1: K pair {2,3}
  int row  = tm * 16 + mn;
  int col  = tn * 16 + mn;

  const float* xp = X + (size_t)row * K + half * 2;  // 8B aligned -> b64 loads
  const float* wb = W + col;

  v8f acc = {};
#pragma unroll
  for (int k0 = 0; k0 < K; k0 += 4) {
    v2f a = *(const v2f*)(xp + k0);
    v2f b;
    b.x = wb[(k0 + half * 2 + 0) * NOUT];
    b.y = wb[(k0 + half * 2 + 1) * NOUT];
    acc = __builtin_amdgcn_wmma_f32_16x16x4_f32(false, a, false, b,
                                                (short)0, acc, false, false);
  }
  float bv = bias[col];
  float* yp = Y + (size_t)(tm * 16 + half * 8) * NOUT + tn * 16 + mn;
#pragma unroll
  for (int r = 0; r < 8; ++r) yp[r * NOUT] = acc[r] + bv;
}

// ----------------------- ordered-float atomic max keys ---------------------
__device__ __forceinline__ unsigned f2ord(float f) {
  unsigned b = __float_as_uint(f);
  return (b & 0x80000000u) ? ~b : (b | 0x80000000u);
}
__device__ __forceinline__ float ord2f(unsigned k) {
  unsigned b = (k & 0x80000000u) ? (k & 0x7fffffffu) : ~k;
  return __uint_as_float(b);
}

// --------------------- pass 1: attention logits + seg-max -------------------
// Warp per edge. Lane owns DOUT/32 contiguous channels; lanes 0-15 cover
// head 0 exactly, lanes 16-31 head 1 -> butterfly reduce within each half.
template <int DOUT>
__global__ __launch_bounds__(256) void edge_alpha_kernel(
    const float* __restrict__ q, const float* __restrict__ k,
    const float* __restrict__ We, const float* __restrict__ ea,
    const int* __restrict__ src, const int* __restrict__ dst,
    float* __restrict__ alpha, unsigned* __restrict__ mord, int E) {
  constexpr int VEC = DOUT / 32;
  int e = (blockIdx.x * blockDim.x + threadIdx.x) >> 5;
  if (e >= E) return;
  int lane = threadIdx.x & 31;
  int j0 = lane * VEC;
  int s = src[e], d = dst[e];
  float eav = ea[e];
  const float* kp = k + (size_t)s * DOUT + j0;
  const float* qp = q + (size_t)d * DOUT + j0;
  const float* wp = We + j0;
  float acc = 0.0f;
#pragma unroll
  for (int t = 0; t < VEC; ++t) acc += qp[t] * (kp[t] + eav * wp[t]);
  acc += __shfl_xor(acc, 1, 32);
  acc += __shfl_xor(acc, 2, 32);
  acc += __shfl_xor(acc, 4, 32);
  acc += __shfl_xor(acc, 8, 32);
  if ((lane & 15) == 0) {
    int h = lane >> 4;
    float a = acc * rsqrtf((float)(DOUT / 2));
    alpha[(size_t)e * 2 + h] = a;
    atomicMax(&mord[(size_t)d * 2 + h], f2ord(a));
  }
}

// --------------------- pass 2: exp(alpha - max) + denom ---------------------
__global__ __launch_bounds__(256) void edge_exp_kernel(
    float* __restrict__ alpha, const unsigned* __restrict__ mord,
    const int* __restrict__ dst, float* __restrict__ denom, int E) {
  int e = blockIdx.x * blockDim.x + threadIdx.x;
  if (e >= E) return;
  int d = dst[e];
  float p0 = expf(alpha[(size_t)e * 2 + 0] - ord2f(mord[(size_t)d * 2 + 0]));
  float p1 = expf(alpha[(size_t)e * 2 + 1] - ord2f(mord[(size_t)d * 2 + 1]));
  alpha[(size_t)e * 2 + 0] = p0;
  alpha[(size_t)e * 2 + 1] = p1;
  atomicAdd(&denom[(size_t)d * 2 + 0], p0);
  atomicAdd(&denom[(size_t)d * 2 + 1], p1);
}

// --------------- pass 3: scatter messages into agg (=skip) ------------------
template <int DOUT>
__global__ __launch_bounds__(256) void edge_scatter_kernel(
    const float* __restrict__ v, const float* __restrict__ We,
    const float* __restrict__ ea, const float* __restrict__ p,
    const float* __restrict__ denom, const int* __restrict__ src,
    const int* __restrict__ dst, float* __restrict__ agg, int E) {
  constexpr int VEC = DOUT / 32;
  int e = (blockIdx.x * blockDim.x + threadIdx.x) >> 5;
  if (e >= E) return;
  int lane = threadIdx.x & 31;
  int h = lane >> 4;
  int j0 = lane * VEC;
  int s = src[e], d = dst[e];
  float eav = ea[e];
  float coef = p[(size_t)e * 2 + h] / (denom[(size_t)d * 2 + h] + 1e-16f);
  const float* vp = v + (size_t)s * DOUT + j0;
  const float* wp = We + j0;
  float* ap = agg + (size_t)d * DOUT + j0;
#pragma unroll
  for (int t = 0; t < VEC; ++t)
    atomicAdd(&ap[t], (vp[t] + eav * wp[t]) * coef);
}

// ------------------------------ BN helpers ---------------------------------
__global__ __launch_bounds__(256) void zero_kernel(float* __restrict__ p, int n) {
  int i = blockIdx.x * blockDim.x + threadIdx.x;
  if (i < n) p[i] = 0.0f;
}

__global__ __launch_bounds__(256) void bn_stats_kernel(
    const float* __restrict__ X, float* __restrict__ stats, int N, int Ch) {
  int t  = blockIdx.x * blockDim.x + threadIdx.x;
  int ch = t % Ch;
  int r0 = t / Ch;
  int rs = (gridDim.x * blockDim.x) / Ch;
  float s = 0.0f, s2 = 0.0f;
  for (int r = r0; r < N; r += rs) {
    float v = X[(size_t)r * Ch + ch];
    s += v; s2 += v * v;
  }
  atomicAdd(&stats[ch], s);
  atomicAdd(&stats[Ch + ch], s2);
}

__global__ __launch_bounds__(256) void bn_relu_kernel(
    const float* __restrict__ X, const float* __restrict__ stats,
    const float* __restrict__ g, const float* __restrict__ b,
    float* __restrict__ Y, int N, int Ch) {
  int i = blockIdx.x * blockDim.x + threadIdx.x;
  if (i >= N * Ch) return;
  int ch = i % Ch;
  float invN = 1.0f / (float)N;
  float mu  = stats[ch] * invN;
  float var = stats[Ch + ch] * invN - mu * mu;  // biased variance
  float y = (X[i] - mu) * rsqrtf(var + 1e-5f) * g[ch] + b[ch];
  Y[i] = fmaxf(y, 0.0f);
}

// ------------------------------- host side ---------------------------------
static inline void launch_gemm(const float* X, const float* W, const float* b,
                               float* Y, int Nrows, int K, int Nout,
                               hipStream_t s) {
  int tilesM = Nrows / 16;              // N_NODES % 16 == 0
  int tiles  = tilesM * (Nout / 16);
  int blocks = (tiles + 7) / 8;         // 8 waves / 256-thread block (wave32)
  if (K == 64)
    gemm_bias_wmma<64, 128><<<blocks, 256, 0, s>>>(X, W, b, Y, tilesM);
  else
    gemm_bias_wmma<128, 64><<<blocks, 256, 0, s>>>(X, W, b, Y, tilesM);
}

extern "C" void kernel_launch(void* const* d_in, const int* in_sizes, int n_in,
                              void* d_out, int out_size, void* d_ws,
                              size_t ws_size, hipStream_t stream) {
  (void)in_sizes; (void)n_in; (void)out_size; (void)ws_size;
  const int N = N_NODES, E = N_EDGES;

  const float* x  = (const float*)d_in[0];
  const float* ea = (const float*)d_in[1];
  struct Conv { const float *Wq,*bq,*Wk,*bk,*Wv,*bv,*We,*Ws,*bs; };
  Conv c[4];
  int p = 2;
  for (int i = 0; i < 4; ++i) {
    c[i].Wq = (const float*)d_in[p++]; c[i].bq = (const float*)d_in[p++];
    c[i].Wk = (const float*)d_in[p++]; c[i].bk = (const float*)d_in[p++];
    c[i].Wv = (const float*)d_in[p++]; c[i].bv = (const float*)d_in[p++];
    c[i].We = (const float*)d_in[p++];
    c[i].Ws = (const float*)d_in[p++]; c[i].bs = (const float*)d_in[p++];
  }
  const float *bng[4], *bnb[4];
  for (int i = 0; i < 4; ++i) {
    bng[i] = (const float*)d_in[p++];
    bnb[i] = (const float*)d_in[p++];
  }
  const int* ei  = (const int*)d_in[p++];
  const int* src = ei;
  const int* dst = ei + E;

  // workspace layout (floats)
  float* ws    = (float*)d_ws;
  float* q     = ws;                         // N*128; doubles as agg after pass1
  float* kbuf  = q    + (size_t)N * 128;     // N*128
  float* vbuf  = kbuf + (size_t)N * 128;     // N*128
  float* t0    = vbuf + (size_t)N * 128;     // N*128 ping buffer (h1, h3)
  float* alpha = t0   + (size_t)N * 128;     // E*2 (logits, then exp weights)
  float* denom = alpha + (size_t)E * 2;      // N*2
  unsigned* mord = (unsigned*)(denom + (size_t)N * 2);  // N*2 ordered keys
  float* stats = (float*)(mord + (size_t)N * 2);        // 2*128

  float* xae = (float*)d_out;                // N*64
  float* xde = xae + (size_t)N * 64;         // N*64

  const float* lin[4] = { x,  t0,  xae, t0  };
  float*      lout[4] = { t0, xae, t0,  xde };
  const int    din[4] = { 64, 128, 64, 128 };
  const int    dou[4] = { 128, 64, 128, 64 };

  const int ewarpBlocks = (E * 32 + 255) / 256;  // warp-per-edge kernels
  const int ethrBlocks  = (E + 255) / 256;
  const int zeroLen     = N * 2 + N * 2 + 256;   // denom + mord + stats

  for (int l = 0; l < 4; ++l) {
    int dIn = din[l], dOut = dou[l];
    const Conv& cv = c[l];

    // Q, K, V projections (WMMA fp32)
    launch_gemm(lin[l], cv.Wq, cv.bq, q,    N, dIn, dOut, stream);
    launch_gemm(lin[l], cv.Wk, cv.bk, kbuf, N, dIn, dOut, stream);
    launch_gemm(lin[l], cv.Wv, cv.bv, vbuf, N, dIn, dOut, stream);

    // reset denom / seg-max keys / BN stats (key 0 == ordered -inf)
    zero_kernel<<<(zeroLen + 255) / 256, 256, 0, stream>>>(denom, zeroLen);

    // pass 1: logits + segment max
    if (dOut == 128)
      edge_alpha_kernel<128><<<ewarpBlocks, 256, 0, stream>>>(
          q, kbuf, cv.We, ea, src, dst, alpha, mord, E);
    else
      edge_alpha_kernel<64><<<ewarpBlocks, 256, 0, stream>>>(
          q, kbuf, cv.We, ea, src, dst, alpha, mord, E);

    // pass 2: exp + segment sum
    edge_exp_kernel<<<ethrBlocks, 256, 0, stream>>>(alpha, mord, dst, denom, E);

    // skip connection directly into agg (q buffer is free after pass 1)
    launch_gemm(lin[l], cv.Ws, cv.bs, q, N, dIn, dOut, stream);

    // pass 3: weighted message scatter-add onto skip
    if (dOut == 128)
      edge_scatter_kernel<128><<<ewarpBlocks, 256, 0, stream>>>(
          vbuf, cv.We, ea, alpha, denom, src, dst, q, E);
    else
      edge_scatter_kernel<64><<<ewarpBlocks, 256, 0, stream>>>(
          vbuf, cv.We, ea, alpha, denom, src, dst, q, E);

    // BatchNorm (training stats) + ReLU
    bn_stats_kernel<<<128, 256, 0, stream>>>(q, stats, N, dOut);
    int nel = N * dOut;
    bn_relu_kernel<<<(nel + 255) / 256, 256, 0, stream>>>(
        q, stats, bng[l], bnb[l], lout[l], N, dOut);
  }
}